// FINDER_net_31112743092387
// MI455X (gfx1250) — compile-verified
//
#include <hip/hip_runtime.h>

#define NN   100000
#define BB   512
#define EE   1000000
#define ESZ  1200000
#define DD   64
#define NBT  (NN + BB)

// workspace layout (float offsets)
#define WS_A0   0
#define WS_A1   (NBT)
#define WS_M0   (2*NBT)
#define WS_M1   (3*NBT)
#define WS_DEG0 (4*NBT)
#define WS_DEG1 (5*NBT)
#define WS_SM   (6*NBT)
#define WS_U1   (WS_SM + 128)
#define WS_U2   (WS_SM + 192)
#define WS_T    (WS_SM + 256)          // 2*512 scalars (ycur . cross_product)
#define WS_ESA  (WS_T + 2*BB)          // 2*512*64
#define WS_HID  (WS_ESA + 2*BB*DD)     // 2*512*32

typedef float v2f __attribute__((ext_vector_type(2)));
typedef float v8f __attribute__((ext_vector_type(8)));

// ---------------- zero accumulators ----------------
__global__ void k_zero(float* ws) {
    int i = blockIdx.x * blockDim.x + threadIdx.x;
    if (i < 6 * NBT) ws[i] = 0.0f;
}

// ---------------- tiny dense chain: v, w, u1, u2 ----------------
__global__ void k_small(const float* __restrict__ w_n2l,
                        const float* __restrict__ p_conv,
                        const float* __restrict__ W_sage,
                        float* ws) {
    __shared__ float v[DD], w[DD];
    int d = threadIdx.x;
    float x = fmaxf(w_n2l[d] + w_n2l[DD + d], 0.0f);   // ones([.,2]) @ w_n2l, relu
    v[d] = x;
    __syncthreads();
    float ss = 0.f;
    for (int k = 0; k < DD; ++k) ss += v[k] * v[k];
    float nv = x / fmaxf(sqrtf(ss), 1e-12f);           // l2 normalize
    __syncthreads();
    v[d] = nv;
    __syncthreads();
    float wd = 0.f;
    for (int k = 0; k < DD; ++k) wd += v[k] * p_conv[k * DD + d];
    w[d] = wd;
    __syncthreads();
    float u1 = 0.f, u2 = 0.f;
    for (int k = 0; k < DD; ++k) {
        u1 += w[k] * W_sage[k * DD + d];
        u2 += w[k] * W_sage[(DD + k) * DD + d];
    }
    ws[WS_SM + d]  = nv;
    ws[WS_SM + 64 + d] = wd;
    ws[WS_U1 + d] = u1;
    ws[WS_U2 + d] = u2;
}

// ---------------- n2n edge value segment sums -> a0/a1 (real nodes) ----------------
__global__ void k_n2n(const int* __restrict__ r0, const float* __restrict__ v0,
                      const int* __restrict__ r1, const float* __restrict__ v1,
                      float* ws) {
    int e = blockIdx.x * blockDim.x + threadIdx.x;
    if (e < EE) {
        atomicAdd(&ws[WS_A0 + r0[e]], v0[e]);
    } else if (e < 2 * EE) {
        int i = e - EE;
        atomicAdd(&ws[WS_A1 + r1[i]], v1[i]);
    }
}

// ---------------- subgraph histogram -> a0/a1 (virtual nodes) ----------------
__global__ void k_subg(const int* __restrict__ subg_row, float* ws) {
    int i = blockIdx.x * blockDim.x + threadIdx.x;
    if (i < NN) {
        int b = subg_row[i];
        atomicAdd(&ws[WS_A0 + NN + b], 1.0f);
        atomicAdd(&ws[WS_A1 + NN + b], 1.0f);
    }
}

// ---------------- SAGE mean-agg numerators + degrees ----------------
__global__ void k_sage(const int* __restrict__ s0r, const int* __restrict__ s0c,
                       const int* __restrict__ s1r, const int* __restrict__ s1c,
                       float* ws) {
    int e = blockIdx.x * blockDim.x + threadIdx.x;
    if (e < ESZ) {
        int r = s0r[e];
        atomicAdd(&ws[WS_M0 + r], ws[WS_A0 + s0c[e]]);
        atomicAdd(&ws[WS_DEG0 + r], 1.0f);
    } else if (e < 2 * ESZ) {
        int i = e - ESZ;
        int r = s1r[i];
        atomicAdd(&ws[WS_M1 + r], ws[WS_A1 + s1c[i]]);
        atomicAdd(&ws[WS_DEG1 + r], 1.0f);
    }
}

// ---------------- rows: msg = relu(a*u1 + m*u2), row l2-norm, write cur / t ----
// one wave per (layer, node) row; each lane owns 2 consecutive d -> float2 store
__global__ void k_rows(float* ws, const float* __restrict__ cp, float* out) {
    int wid  = (blockIdx.x * blockDim.x + threadIdx.x) >> 5;
    int lane = threadIdx.x & 31;
    if (wid >= 2 * NBT) return;
    int l = wid / NBT;
    int i = wid - l * NBT;

    float a   = ws[(l ? WS_A1  : WS_A0)  + i];
    float num = ws[(l ? WS_M1  : WS_M0)  + i];
    float dg  = ws[(l ? WS_DEG1: WS_DEG0)+ i];
    float m   = num / fmaxf(dg, 1.0f);

    int d0 = 2 * lane, d1 = 2 * lane + 1;
    float r0 = fmaxf(a * ws[WS_U1 + d0] + m * ws[WS_U2 + d0], 0.0f);
    float r1 = fmaxf(a * ws[WS_U1 + d1] + m * ws[WS_U2 + d1], 0.0f);

    float ss = r0 * r0 + r1 * r1;
    for (int off = 16; off >= 1; off >>= 1) ss += __shfl_xor(ss, off, 32);
    float inv = 1.0f / fmaxf(sqrtf(ss), 1e-12f);
    r0 *= inv; r1 *= inv;

    if (i < NN) {
        float2 pr = make_float2(r0, r1);
        float* dst = out + BB + (l * NN + i) * DD + d0;   // cur output region
        *(float2*)dst = pr;
    } else {
        int b = i - NN;
        float t = r0 * cp[d0] + r1 * cp[d1];              // ycur . cross_product
        for (int off = 16; off >= 1; off >>= 1) t += __shfl_xor(t, off, 32);
        if (lane == 0) ws[WS_T + l * BB + b] = t;
    }
}

// ---------------- esa[l][b][d] = t[l][b] * cur[l][act_col[b]][d] ----------------
__global__ void k_esa(const int* __restrict__ act_col, const float* __restrict__ out,
                      float* ws) {
    int idx = blockIdx.x * blockDim.x + threadIdx.x;     // 2*512*64
    if (idx >= 2 * BB * DD) return;
    int d = idx & (DD - 1);
    int b = (idx >> 6) & (BB - 1);
    int l = idx >> 15;
    float t = ws[WS_T + l * BB + b];
    float ae = out[BB + (l * NN + act_col[b]) * DD + d];
    ws[WS_ESA + idx] = t * ae;
}

// ---------------- WMMA GEMM: hid = relu(esa[2*512,64] @ h1[64,32]) ----------------
// one wave per 16x16 tile; V_WMMA_F32_16X16X4_F32, 16 k-steps of K=4
__global__ void k_gemm(const float* __restrict__ ws, const float* __restrict__ h1,
                       float* __restrict__ hid) {
    int bid  = blockIdx.x;          // 128 = 2 layers * 32 M-tiles * 2 N-tiles
    int tn   = bid & 1;
    int tm   = (bid >> 1) & 31;
    int l    = bid >> 6;
    int lane = threadIdx.x;
    int half = lane >> 4;           // 0: lanes 0-15, 1: lanes 16-31
    int lid  = lane & 15;

    const float* A = ws + WS_ESA + (l * BB + tm * 16) * DD;  // 16 x 64 tile
    v8f c = {0.f, 0.f, 0.f, 0.f, 0.f, 0.f, 0.f, 0.f};
    #pragma unroll
    for (int s = 0; s < 16; ++s) {
        int k0 = 4 * s + 2 * half;
        v2f av, bv;
        av.x = A[lid * DD + k0];
        av.y = A[lid * DD + k0 + 1];
        bv.x = h1[(k0    ) * 32 + tn * 16 + lid];
        bv.y = h1[(k0 + 1) * 32 + tn * 16 + lid];
        c = __builtin_amdgcn_wmma_f32_16x16x4_f32(false, av, false, bv,
                                                  (short)0, c, false, false);
    }
    // C layout: VGPR r holds M=r (lanes 0-15) and M=r+8 (lanes 16-31), N=lid
    float* dst = hid + (l * BB + tm * 16) * 32 + tn * 16;
    #pragma unroll
    for (int r = 0; r < 8; ++r) {
        int M = r + 8 * half;
        dst[M * 32 + lid] = fmaxf(c[r], 0.0f);
    }
}

// ---------------- q[b] = sum_l ( hid[l][b]·h2w[0:32] + aux[b][l]·h2w[32:36] ) ----
__global__ void k_q(const float* __restrict__ ws, const float* __restrict__ aux,
                    const float* __restrict__ h2w, float* out) {
    int b = blockIdx.x * blockDim.x + threadIdx.x;
    if (b >= BB) return;
    float q = 0.0f;
    for (int l = 0; l < 2; ++l) {
        const float* h = ws + WS_HID + (l * BB + b) * 32;
        for (int j = 0; j < 32; ++j) q += h[j] * h2w[j];
        for (int j = 0; j < 4; ++j)  q += aux[b * 8 + l * 4 + j] * h2w[32 + j];
    }
    out[b] = q;
}

extern "C" void kernel_launch(void* const* d_in, const int* in_sizes, int n_in,
                              void* d_out, int out_size, void* d_ws, size_t ws_size,
                              hipStream_t stream) {
    const int*   n2n0_row = (const int*)  d_in[0];
    const float* n2n0_val = (const float*)d_in[2];
    const int*   n2n1_row = (const int*)  d_in[3];
    const float* n2n1_val = (const float*)d_in[5];
    const int*   subg_row = (const int*)  d_in[6];
    const int*   act_col  = (const int*)  d_in[9];
    const int*   sg0_row  = (const int*)  d_in[10];
    const int*   sg0_col  = (const int*)  d_in[11];
    const int*   sg1_row  = (const int*)  d_in[12];
    const int*   sg1_col  = (const int*)  d_in[13];
    const float* aux      = (const float*)d_in[14];
    const float* w_n2l    = (const float*)d_in[15];
    const float* p_conv   = (const float*)d_in[16];
    const float* W_sage   = (const float*)d_in[17];
    const float* cp       = (const float*)d_in[18];
    const float* h1w      = (const float*)d_in[19];
    const float* h2w      = (const float*)d_in[20];
    float* out = (float*)d_out;
    float* ws  = (float*)d_ws;

    k_zero <<<(6 * NBT + 255) / 256, 256, 0, stream>>>(ws);
    k_small<<<1, 64, 0, stream>>>(w_n2l, p_conv, W_sage, ws);
    k_n2n  <<<(2 * EE + 255) / 256, 256, 0, stream>>>(n2n0_row, n2n0_val,
                                                      n2n1_row, n2n1_val, ws);
    k_subg <<<(NN + 255) / 256, 256, 0, stream>>>(subg_row, ws);
    k_sage <<<(2 * ESZ + 255) / 256, 256, 0, stream>>>(sg0_row, sg0_col,
                                                       sg1_row, sg1_col, ws);
    k_rows <<<(2 * NBT * 32 + 255) / 256, 256, 0, stream>>>(ws, cp, out);
    k_esa  <<<(2 * BB * DD + 255) / 256, 256, 0, stream>>>(act_col, out, ws);
    k_gemm <<<128, 32, 0, stream>>>(ws, h1w, ws + WS_HID);
    k_q    <<<(BB + 255) / 256, 256, 0, stream>>>(ws, aux, h2w, out);
}